// WSGATLayer_10093173145802
// MI455X (gfx1250) — compile-verified
//
#include <hip/hip_runtime.h>

// ---------------------------------------------------------------------------
// WSGAT layer for MI455X (gfx1250, wave32).
//   k_gemm_attn : z = h @ W_fc via V_WMMA_F32_16X16X4_F32; one wave computes a
//                 full 16x64 tile (4 accumulators, A fragment reused 4x), then
//                 fuses the attention-logit epilogue:
//                 e_node[w] = mask0(leaky_relu(z_w . W_attn[0:64]))
//                 via lane-local FMA + 16-lane __shfl_xor reduction.
//   k_init      : zero sh & denom, segmax = -inf
//   k_segmax    : per-edge f32 max atomic into segmax[dst]
//   k_fixmax    : empty segments: -inf -> 0
//   k_exp       : ex_e = exp(e[src]-segmax[dst]); f32 add atomic into denom
//   k_agg       : sh[dst] += (ex/denom) * z[src]  (wave per edge, 64 feats)
// Dead code in reference (ws_embed @ W_feat) is intentionally skipped.
// ---------------------------------------------------------------------------

typedef float v2f __attribute__((ext_vector_type(2)));
typedef float v8f __attribute__((ext_vector_type(8)));

#define IN_DIM    128
#define OUT_DIM   64
#define NEG_SLOPE 0.01f

// -------- GEMM + fused attention logit: one wave per 16-row stripe ---------
__global__ __launch_bounds__(256) void k_gemm_attn(const float* __restrict__ h,
                                                   const float* __restrict__ W_fc,
                                                   const float* __restrict__ W_attn,
                                                   float* __restrict__ z,
                                                   float* __restrict__ e_node,
                                                   int n_word) {
  const int tm   = (blockIdx.x * blockDim.x + threadIdx.x) >> 5;  // tile row
  const int lane = threadIdx.x & 31;
  if (tm * 16 >= n_word) return;               // wave-uniform (M % 16 == 0)

  const int half = lane >> 4;                  // 0: K pair {0,1}; 1: {2,3}
  const int r    = lane & 15;

  const float* __restrict__ arow  = h + (size_t)(tm * 16 + r) * IN_DIM + 2 * half;
  const float* __restrict__ bbase = W_fc + r;

  v8f c0 = {0.f,0.f,0.f,0.f,0.f,0.f,0.f,0.f};
  v8f c1 = c0, c2 = c0, c3 = c0;

#pragma unroll
  for (int k = 0; k < IN_DIM; k += 4) {
    // A 16x4 f32 fragment: lane(r,half) holds A[r][k+2h], A[r][k+2h+1]
    v2f a;
    a.x = arow[k];
    a.y = arow[k + 1];
    // B 4x16 fragments for the four 16-col tiles; A reused 4x
    const float* __restrict__ b0 = bbase + (size_t)(k + 2 * half)     * OUT_DIM;
    const float* __restrict__ b1 = bbase + (size_t)(k + 2 * half + 1) * OUT_DIM;
    v2f b;
    b.x = b0[0];  b.y = b1[0];
    c0 = __builtin_amdgcn_wmma_f32_16x16x4_f32(false, a, false, b, (short)0, c0, false, false);
    b.x = b0[16]; b.y = b1[16];
    c1 = __builtin_amdgcn_wmma_f32_16x16x4_f32(false, a, false, b, (short)0, c1, false, false);
    b.x = b0[32]; b.y = b1[32];
    c2 = __builtin_amdgcn_wmma_f32_16x16x4_f32(false, a, false, b, (short)0, c2, false, false);
    b.x = b0[48]; b.y = b1[48];
    c3 = __builtin_amdgcn_wmma_f32_16x16x4_f32(false, a, false, b, (short)0, c3, false, false);
  }

  // D 16x16 f32: VGPR i -> row i (lanes 0-15) / row i+8 (lanes 16-31), col = r
  float* __restrict__ zrow =
      z + (size_t)(tm * 16 + half * 8) * OUT_DIM + r;
#pragma unroll
  for (int i = 0; i < 8; ++i) {
    zrow[(size_t)i * OUT_DIM +  0] = c0[i];
    zrow[(size_t)i * OUT_DIM + 16] = c1[i];
    zrow[(size_t)i * OUT_DIM + 32] = c2[i];
    zrow[(size_t)i * OUT_DIM + 48] = c3[i];
  }

  // Fused attention logit: t[row] = z[row] . W_attn[0:64]
  // lane-local partial over its 4 columns, then 16-lane xor-butterfly
  // (masks 1,2,4,8 keep the two half-wave groups independent).
  const float w0 = W_attn[r];
  const float w1 = W_attn[16 + r];
  const float w2 = W_attn[32 + r];
  const float w3 = W_attn[48 + r];
#pragma unroll
  for (int i = 0; i < 8; ++i) {
    float p = c0[i] * w0 + c1[i] * w1 + c2[i] * w2 + c3[i] * w3;
    p += __shfl_xor(p, 1);
    p += __shfl_xor(p, 2);
    p += __shfl_xor(p, 4);
    p += __shfl_xor(p, 8);
    if (r == 0) {                                 // lanes 0 and 16 write
      float s = p > 0.f ? p : NEG_SLOPE * p;      // leaky_relu
      e_node[tm * 16 + half * 8 + i] = (s == 0.f) ? -1000.f : s;
    }
  }
}

// ---------------- init output / denom / segmax -----------------------------
__global__ void k_init(float* __restrict__ sh, float* __restrict__ denom,
                       float* __restrict__ segmax, int n_sent) {
  int i = blockIdx.x * blockDim.x + threadIdx.x;
  if (i < n_sent * OUT_DIM) sh[i] = 0.f;
  if (i < n_sent) {
    denom[i]  = 0.f;
    segmax[i] = -__builtin_inff();
  }
}

// ---------------- segment max over edges -----------------------------------
__global__ void k_segmax(const float* __restrict__ e_node,
                         const int* __restrict__ edge_src,
                         const int* __restrict__ edge_dst,
                         float* __restrict__ segmax, int E) {
  int e = blockIdx.x * blockDim.x + threadIdx.x;
  if (e >= E) return;
  unsafeAtomicMax(&segmax[edge_dst[e]], e_node[edge_src[e]]);
}

__global__ void k_fixmax(float* __restrict__ segmax, int n_sent) {
  int i = blockIdx.x * blockDim.x + threadIdx.x;
  if (i >= n_sent) return;
  float m = segmax[i];
  if (m == -__builtin_inff()) segmax[i] = 0.f;   // empty segment
}

// ---------------- exp + segment sum ----------------------------------------
__global__ void k_exp(const float* __restrict__ e_node,
                      const int* __restrict__ edge_src,
                      const int* __restrict__ edge_dst,
                      const float* __restrict__ segmax,
                      float* __restrict__ ex, float* __restrict__ denom, int E) {
  int e = blockIdx.x * blockDim.x + threadIdx.x;
  if (e >= E) return;
  int d = edge_dst[e];
  float v = expf(e_node[edge_src[e]] - segmax[d]);
  ex[e] = v;
  unsafeAtomicAdd(&denom[d], v);
}

// ---------------- weighted aggregation: wave per edge ----------------------
__global__ __launch_bounds__(256) void k_agg(const float* __restrict__ z,
                                             const float* __restrict__ ex,
                                             const float* __restrict__ denom,
                                             const int* __restrict__ edge_src,
                                             const int* __restrict__ edge_dst,
                                             float* __restrict__ sh, int E) {
  int e    = (blockIdx.x * blockDim.x + threadIdx.x) >> 5;
  int lane = threadIdx.x & 31;
  if (e >= E) return;
  int s = edge_src[e];
  int d = edge_dst[e];
  float den   = denom[d];
  float alpha = ex[e] / (den == 0.f ? 1.f : den);
  const float* __restrict__ zr = z + (size_t)s * OUT_DIM;
  float* __restrict__ o        = sh + (size_t)d * OUT_DIM;
  unsafeAtomicAdd(&o[lane],      alpha * zr[lane]);
  unsafeAtomicAdd(&o[lane + 32], alpha * zr[lane + 32]);
}

// ---------------------------------------------------------------------------
extern "C" void kernel_launch(void* const* d_in, const int* in_sizes, int n_in,
                              void* d_out, int out_size, void* d_ws, size_t ws_size,
                              hipStream_t stream) {
  const float* h        = (const float*)d_in[0];
  const float* ws_embed = (const float*)d_in[1];  // dead in reference
  const float* W_fc     = (const float*)d_in[2];
  const float* W_feat   = (const float*)d_in[3];  // dead in reference
  const float* W_attn   = (const float*)d_in[4];
  const int*   edge_src = (const int*)d_in[5];
  const int*   edge_dst = (const int*)d_in[6];
  float*       sh       = (float*)d_out;
  (void)ws_embed; (void)W_feat; (void)n_in; (void)ws_size;

  const int n_word = in_sizes[0] / IN_DIM;   // 200000
  const int E      = in_sizes[5];            // 800000
  const int n_sent = out_size / OUT_DIM;     // 20000

  // workspace carve-up (~55.4 MB)
  float* ws     = (float*)d_ws;
  float* z      = ws;                              // n_word*64
  float* e_node = z + (size_t)n_word * OUT_DIM;    // n_word
  float* segmax = e_node + n_word;                 // n_sent
  float* denom  = segmax + n_sent;                 // n_sent
  float* ex     = denom + n_sent;                  // E

  // 1) GEMM + fused attention logits: one wave per 16x64 tile of z
  {
    int waves  = (n_word + 15) / 16;
    int blocks = (waves * 32 + 255) / 256;
    k_gemm_attn<<<blocks, 256, 0, stream>>>(h, W_fc, W_attn, z, e_node, n_word);
  }
  // 2) init accumulators
  k_init<<<(n_sent * OUT_DIM + 255) / 256, 256, 0, stream>>>(sh, denom, segmax, n_sent);
  // 3) segment max
  k_segmax<<<(E + 255) / 256, 256, 0, stream>>>(e_node, edge_src, edge_dst, segmax, E);
  k_fixmax<<<(n_sent + 255) / 256, 256, 0, stream>>>(segmax, n_sent);
  // 4) exp + segment sum
  k_exp<<<(E + 255) / 256, 256, 0, stream>>>(e_node, edge_src, edge_dst, segmax, ex, denom, E);
  // 5) weighted gather-scatter aggregation (wave per edge)
  k_agg<<<((size_t)E * 32 + 255) / 256, 256, 0, stream>>>(z, ex, denom, edge_src, edge_dst, sh, E);
}